// IntraAgg_5239860101744
// MI455X (gfx1250) — compile-verified
//
#include <hip/hip_runtime.h>

#define K_NBR 32
#define D_DIM 256

// Kernel 1: invert node-id -> embedding-row mapping: inv[u[i]] = i
__global__ void build_inv_kernel(const int* __restrict__ uidx,
                                 int* __restrict__ inv, int n) {
    int i = blockIdx.x * blockDim.x + threadIdx.x;
    if (i < n) inv[uidx[i]] = i;
}

// Kernel 2: one 256-thread block (8 wave32) per batch row.
//   Phase A : async-gather the 32 neighbor rows (1 KB each) into LDS via
//             global_load_async_to_lds_b128 (ASYNCcnt-tracked, no VGPR staging).
//   Phase A2: wave 0 computes the distinct-neighbor keep mask (wave32 == K).
//   Phase B : branchless per-column masked mean, self-loop subtract, concat.
__global__ __launch_bounds__(256) void intra_agg_kernel(
    const float* __restrict__ embedding,
    const int*   __restrict__ nbr,
    const int*   __restrict__ inv,
    const float* __restrict__ self_feats,
    float*       __restrict__ out)
{
    __shared__ float    s_emb[K_NBR * D_DIM];  // 32 KB staged neighbor rows
    __shared__ unsigned s_mask;                // distinct-neighbor keep bits
    __shared__ int      s_cnt;                 // popcount(s_mask)

    const int b = blockIdx.x;
    const int t = threadIdx.x;

    // ---- Phase A: issue async gathers; thread t owns neighbor k = t/8 and
    //      a contiguous 128-byte slice (8 x b128) of that 1 KB row. ----
    {
        const int k   = t >> 3;
        const int sub = (t & 7) * 128;                 // byte offset in row
        int id  = nbr[b * K_NBR + k];
        int idc = id < 0 ? 0 : id;                     // clip like reference
        int row = inv[idc];
        const char* gsrc =
            (const char*)(embedding + (size_t)row * D_DIM) + sub;
        // Low 32 bits of a generic LDS pointer = LDS byte offset.
        unsigned lds =
            (unsigned)(unsigned long long)(&s_emb[k * D_DIM]) + (unsigned)sub;
#pragma unroll
        for (int i = 0; i < 8; ++i) {
            asm volatile("global_load_async_to_lds_b128 %0, %1, off"
                         :: "v"(lds + (unsigned)(i * 16)),
                            "v"(gsrc + i * 16)
                         : "memory");
        }
    }

    // ---- Phase A2: wave 0, one lane per neighbor slot; keep first
    //      occurrence of each distinct valid id (== sorted-dedup set). ----
    if (t < K_NBR) {
        int id = nbr[b * K_NBR + t];
        bool valid = (id >= 0);
        bool dup = false;
#pragma unroll
        for (int j = 0; j < K_NBR; ++j) {
            int other = __shfl(id, j, 32);
            if (j < t && other == id) dup = true;
        }
        bool keep = valid && !dup;
        unsigned long long m = __ballot(keep);
        if (t == 0) {
            s_mask = (unsigned)m;
            s_cnt  = __popcll(m);
        }
    }

    // Each wave drains its own async gathers, then block-wide barrier.
    asm volatile("s_wait_asynccnt 0" ::: "memory");
    __syncthreads();

    // ---- Phase B: thread t reduces column d = t across the kept LDS rows.
    //      Branchless: fold each mask bit with a select-add (0/1 weights make
    //      this exactly the reference's masked mean). Consecutive lanes hit
    //      consecutive LDS banks: conflict-free. ----
    const unsigned mask = s_mask;
    const float    cnt  = (float)s_cnt;

    float acc = 0.0f;
#pragma unroll
    for (int k = 0; k < K_NBR; ++k) {
        float v = s_emb[k * D_DIM + t];
        acc += (mask & (1u << k)) ? v : 0.0f;
    }

    float f1 = acc / cnt;
    const size_t ob = (size_t)b * (2 * D_DIM);
    out[ob + t]         = f1;
    out[ob + D_DIM + t] = self_feats[(size_t)b * D_DIM + t] - f1;
}

extern "C" void kernel_launch(void* const* d_in, const int* in_sizes, int n_in,
                              void* d_out, int out_size, void* d_ws, size_t ws_size,
                              hipStream_t stream) {
    // setup_inputs order: embedding, nodes, neighbor_lists,
    //                     unique_nodes_new_index, self_feats
    const float* embedding      = (const float*)d_in[0];
    const int*   neighbor_lists = (const int*)d_in[2];
    const int*   uidx           = (const int*)d_in[3];
    const float* self_feats     = (const float*)d_in[4];
    float*       out            = (float*)d_out;

    const int n_nodes = in_sizes[3];            // 100000
    const int B       = in_sizes[2] / K_NBR;    // 8192

    int* inv = (int*)d_ws;                      // 400 KB scratch

    build_inv_kernel<<<(n_nodes + 255) / 256, 256, 0, stream>>>(uidx, inv, n_nodes);
    intra_agg_kernel<<<B, 256, 0, stream>>>(embedding, neighbor_lists, inv,
                                            self_feats, out);
}